// Loss_3040836845617
// MI455X (gfx1250) — compile-verified
//
#include <hip/hip_runtime.h>
#include <math.h>

typedef __attribute__((ext_vector_type(2))) float v2f;
typedef __attribute__((ext_vector_type(8))) float v8f;

#define NB      4
#define NPTS    4096
#define NPOINT  204            // int(4096*0.05)
#define REP_H   0.0005f
#define BIGF    3.0e38f

#define CHUNK_PTS 512
#define NCHUNK    (NPTS / CHUNK_PTS)          // 8
#define B128_PER_CHUNK ((CHUNK_PTS * 12) / (32 * 16))   // 12 instrs (6KB / 512B)

// ---------------------------------------------------------------- utilities

// Branchless sorted-ascending top-5 insertion: 9 v_min/v_max, no exec churn.
__device__ __forceinline__ void insert5(float t[5], float v) {
    float a = v;
    float n0 = fminf(t[0], a); a = fmaxf(t[0], a); t[0] = n0;
    float n1 = fminf(t[1], a); a = fmaxf(t[1], a); t[1] = n1;
    float n2 = fminf(t[2], a); a = fmaxf(t[2], a); t[2] = n2;
    float n3 = fminf(t[3], a); a = fmaxf(t[3], a); t[3] = n3;
    t[4] = fminf(t[4], a);
}

// D = A(16x4) x B(4x16) + C, f32, one wave.
__device__ __forceinline__ v8f wmma_dot(v2f a, v2f b, v8f c) {
    return __builtin_amdgcn_wmma_f32_16x16x4_f32(
        false, a, false, b, (short)0, c, false, false);
}

// ---------------------------------------------------------------- init

__global__ __launch_bounds__(64) void init_kernel(float* acc) {
    if (threadIdx.x < 16) acc[threadIdx.x] = 0.0f;
}

// ---------------------------------------------------------------- FPS

__global__ __launch_bounds__(256) void fps_kernel(const float* __restrict__ pcd,
                                                  float* __restrict__ newxyz) {
    const int b   = blockIdx.x;
    const int tid = threadIdx.x;
    const float* P = pcd + (size_t)b * NPTS * 3;

    __shared__ float mind[NPTS];
    __shared__ float rv[256];
    __shared__ int   ri[256];
    __shared__ int   s_last;

    for (int i = tid; i < NPTS; i += 256) mind[i] = BIGF;
    if (tid == 0) s_last = 0;
    __syncthreads();

    for (int k = 0; k < NPOINT; k++) {
        const int last = s_last;
        const float px = P[last * 3 + 0];
        const float py = P[last * 3 + 1];
        const float pz = P[last * 3 + 2];
        if (tid == 0) {
            float* o = newxyz + ((size_t)b * NPOINT + k) * 3;
            o[0] = px; o[1] = py; o[2] = pz;
        }
        float bv = -1.0f; int bi = 0;
        for (int i = tid; i < NPTS; i += 256) {
            const float dx = P[i * 3 + 0] - px;
            const float dy = P[i * 3 + 1] - py;
            const float dz = P[i * 3 + 2] - pz;
            const float d  = dx * dx + dy * dy + dz * dz;
            const float m  = fminf(mind[i], d);
            mind[i] = m;
            if (m > bv) { bv = m; bi = i; }
        }
        rv[tid] = bv; ri[tid] = bi;
        __syncthreads();
        for (int s = 128; s > 0; s >>= 1) {
            if (tid < s) {
                const float v2 = rv[tid + s]; const int i2 = ri[tid + s];
                if (v2 > rv[tid] || (v2 == rv[tid] && i2 < ri[tid])) { rv[tid] = v2; ri[tid] = i2; }
            }
            __syncthreads();
        }
        if (tid == 0) s_last = ri[0];
        __syncthreads();
    }
}

// ---------------------------------------------------------------- repulsion

// One wave per (batch, 16-row tile). Column points are pipelined into LDS with
// GLOBAL_LOAD_ASYNC_TO_LDS_B128 (double buffered, ASYNCcnt-tracked); distances
// via WMMA 16x16x4 f32; branchless per-lane top-5; LDS lane merge.
__global__ __launch_bounds__(32) void repulsion_kernel(const float* __restrict__ pcd,
                                                       float* __restrict__ acc_rep) {
    const int blk  = blockIdx.x;
    const int b    = blk / (NPTS / 16);
    const int rt   = blk % (NPTS / 16);
    const int lane = threadIdx.x;
    const int col  = lane & 15;
    const bool hi  = lane >= 16;
    const float* P = pcd + (size_t)b * NPTS * 3;

    __shared__ float sbuf[2][CHUNK_PTS * 3];     // 2 x 6KB interleaved xyz
    __shared__ float rn[16];
    __shared__ float cand[16][16][5];
    __shared__ float red[32];

    // A matrix (16x4 f32): lanes 0-15 -> K=0,1 (x,y); lanes 16-31 -> K=2,3 (z,0)
    const int r = rt * 16 + col;
    const float ax = P[r * 3 + 0], ay = P[r * 3 + 1], az = P[r * 3 + 2];
    v2f A; A.x = hi ? az : ax; A.y = hi ? 0.0f : ay;
    if (lane < 16) rn[lane] = ax * ax + ay * ay + az * az;
    __syncthreads();

    float rn8[8];
#pragma unroll
    for (int i = 0; i < 8; i++) rn8[i] = rn[(hi ? 8 : 0) + i];

    float t5[8][5];
#pragma unroll
    for (int i = 0; i < 8; i++)
#pragma unroll
        for (int k = 0; k < 5; k++) t5[i][k] = BIGF;

    // async-stage one 512-point chunk (12 x b128 per wave)
    auto issue_chunk = [&](int chunk, int bufidx) {
        const unsigned lbase =
            (unsigned)(unsigned long long)(const void*)&sbuf[bufidx][0];
        const unsigned long long gbase =
            (unsigned long long)(const void*)(P + (size_t)chunk * CHUNK_PTS * 3);
#pragma unroll
        for (int j = 0; j < B128_PER_CHUNK; j++) {
            const unsigned laddr = lbase + (unsigned)(lane * 16 + j * 512);
            const unsigned long long gaddr = gbase + (unsigned long long)(lane * 16 + j * 512);
            asm volatile("global_load_async_to_lds_b128 %0, %1, off"
                         :: "v"(laddr), "v"(gaddr) : "memory");
        }
    };

    issue_chunk(0, 0);
    for (int chunk = 0; chunk < NCHUNK; chunk++) {
        if (chunk + 1 < NCHUNK) {
            issue_chunk(chunk + 1, (chunk + 1) & 1);
            // in-order completion: <=12 outstanding => previous chunk landed
            asm volatile("s_wait_asynccnt 0xc" ::: "memory");
        } else {
            asm volatile("s_wait_asynccnt 0x0" ::: "memory");
        }
        const float* p = &sbuf[chunk & 1][0];
        for (int t = 0; t < CHUNK_PTS / 16; t++) {
            const int cc = t * 16 + col;
            const float bx = p[cc * 3 + 0], by = p[cc * 3 + 1], bz = p[cc * 3 + 2];
            v2f Bm; Bm.x = hi ? bz : bx; Bm.y = hi ? 0.0f : by;
            v8f C = {};
            C = wmma_dot(A, Bm, C);
            const float cn = bx * bx + by * by + bz * bz;
#pragma unroll
            for (int i = 0; i < 8; i++) {
                const float d2 = fmaxf(rn8[i] + cn - 2.0f * C[i], 0.0f);
                insert5(t5[i], d2);
            }
        }
    }

    // merge: row m candidates live in 16 lanes (rows 0-7 <- lanes 0-15, 8-15 <- 16-31)
#pragma unroll
    for (int i = 0; i < 8; i++)
#pragma unroll
        for (int k = 0; k < 5; k++) cand[(hi ? 8 : 0) + i][col][k] = t5[i][k];
    __syncthreads();

    float part = 0.0f;
    if (lane < 16) {
        float best[5] = {BIGF, BIGF, BIGF, BIGF, BIGF};
        for (int l = 0; l < 16; l++)
#pragma unroll
            for (int k = 0; k < 5; k++) insert5(best, cand[lane][l][k]);
        // best[0] = self (0); neighbors 1..4: dist = d2^2, relu(h - dist)
#pragma unroll
        for (int k = 1; k < 5; k++) {
            const float d = best[k] * best[k];
            part += fmaxf(REP_H - d, 0.0f);
        }
    }
    red[lane] = (lane < 16) ? part : 0.0f;
    __syncthreads();
    if (lane == 0) {
        float s = 0.0f;
        for (int i = 0; i < 16; i++) s += red[i];
        atomicAdd(acc_rep, s);
    }
}

// ---------------------------------------------------------------- ball query + uniform

// One wave per group; K/KT compile-time => fully unrolled WMMA tile loops.
template <int K, int KT>
__global__ __launch_bounds__(32) void group_kernel(const float* __restrict__ pcd,
                                                   const float* __restrict__ newxyz,
                                                   float* __restrict__ accp,
                                                   float r2) {
    const int g    = blockIdx.x;                 // 0 .. NB*NPOINT-1
    const int b    = g / NPOINT;
    const int c    = g % NPOINT;
    const int lane = threadIdx.x;
    const int col  = lane & 15;
    const bool hi  = lane >= 16;

    __shared__ float gx[64], gy[64], gz[64];
    __shared__ float rnorm[64];
    __shared__ float minbuf[16][16];
    __shared__ float red[32];

    for (int i = lane; i < 64; i += 32) { gx[i] = 0.0f; gy[i] = 0.0f; gz[i] = 0.0f; }

    const float cx = newxyz[((size_t)b * NPOINT + c) * 3 + 0];
    const float cy = newxyz[((size_t)b * NPOINT + c) * 3 + 1];
    const float cz = newxyz[((size_t)b * NPOINT + c) * 3 + 2];
    const float* P = pcd + (size_t)b * NPTS * 3;
    __syncthreads();

    // ordered compaction of first K in-range points (index-ascending, zero-padded)
    int filled = 0;
    for (int base = 0; base < NPTS && filled < K; base += 32) {
        const int idx = base + lane;
        const float px = P[idx * 3 + 0], py = P[idx * 3 + 1], pz = P[idx * 3 + 2];
        const float dx = px - cx, dy = py - cy, dz = pz - cz;
        const float d2 = dx * dx + dy * dy + dz * dz;
        const bool in = d2 < r2;
        const unsigned m32 = (unsigned)__ballot(in);
        const int prefix = __popc(m32 & ((1u << lane) - 1u));
        const int pos = filled + prefix;
        if (in && pos < K) { gx[pos] = px; gy[pos] = py; gz[pos] = pz; }
        filled += __popc(m32);
    }
    __syncthreads();

    for (int i = lane; i < 64; i += 32)
        rnorm[i] = gx[i] * gx[i] + gy[i] * gy[i] + gz[i] * gz[i];
    __syncthreads();

    float rowacc = 0.0f;    // lanes 0..15: sum over their rows of |minother + 1e-8|
#pragma unroll
    for (int rtile = 0; rtile < KT; rtile++) {
        const int ra = rtile * 16 + col;
        v2f A; A.x = hi ? gz[ra] : gx[ra]; A.y = hi ? 0.0f : gy[ra];
        float rn8[8];
#pragma unroll
        for (int i = 0; i < 8; i++) rn8[i] = rnorm[rtile * 16 + (hi ? 8 : 0) + i];

        float mn[8];
#pragma unroll
        for (int i = 0; i < 8; i++) mn[i] = BIGF;

#pragma unroll
        for (int ctile = 0; ctile < KT; ctile++) {
            const int cb = ctile * 16 + col;
            const float bx = gx[cb], by = gy[cb], bz = gz[cb];
            v2f Bm; Bm.x = hi ? bz : bx; Bm.y = hi ? 0.0f : by;
            v8f C = {};
            C = wmma_dot(A, Bm, C);
            const float cn = bx * bx + by * by + bz * bz;
            const int j = ctile * 16 + col;        // global column slot
#pragma unroll
            for (int i = 0; i < 8; i++) {
                const int m = rtile * 16 + (hi ? 8 : 0) + i;
                const float d2 = fmaxf(rn8[i] + cn - 2.0f * C[i], 0.0f);
                const bool valid = (j < K) && (j != m);
                mn[i] = valid ? fminf(mn[i], d2) : mn[i];
            }
        }
#pragma unroll
        for (int i = 0; i < 8; i++) minbuf[(hi ? 8 : 0) + i][col] = mn[i];
        __syncthreads();
        if (lane < 16) {
            const int m = rtile * 16 + lane;
            if (m < K) {
                float v = minbuf[lane][0];
                for (int l = 1; l < 16; l++) v = fminf(v, minbuf[lane][l]);
                rowacc += fabsf(v + 1e-8f);
            }
        }
        __syncthreads();
    }

    red[lane] = (lane < 16) ? rowacc : 0.0f;
    __syncthreads();
    if (lane == 0) {
        float s = 0.0f;
        for (int i = 0; i < 16; i++) s += red[i];
        const float u = s / (float)K;
        const float expect_len = sqrtf(3.14159265358979323846f / (float)NPTS);
        const float d = u - expect_len;
        atomicAdd(accp, d * d / (expect_len + 1e-8f));
    }
}

// ---------------------------------------------------------------- final combine

__global__ __launch_bounds__(32) void final_kernel(const float* __restrict__ acc,
                                                   float* __restrict__ out) {
    if (threadIdx.x == 0) {
        const float w[5] = {0.16f, 0.36f, 0.64f, 1.0f, 1.44f};  // (p*100)^2
        float ul = 0.0f;
        for (int i = 0; i < 5; i++) ul += (acc[i] / (float)(NB * NPOINT)) * w[i];
        ul *= (1.0f / 5.0f);
        const float rep = acc[5] / (float)(NB * NPTS * 4);
        out[0] = ul + rep;
    }
}

// ---------------------------------------------------------------- launch

extern "C" void kernel_launch(void* const* d_in, const int* in_sizes, int n_in,
                              void* d_out, int out_size, void* d_ws, size_t ws_size,
                              hipStream_t stream) {
    (void)in_sizes; (void)n_in; (void)out_size; (void)ws_size;
    const float* pcd = (const float*)d_in[0];           // (4, 4096, 3) f32
    float* acc    = (float*)d_ws;                       // [0..4] uniform per-p, [5] repulsion
    float* newxyz = acc + 16;                           // NB*NPOINT*3 floats
    float* out    = (float*)d_out;

    init_kernel<<<1, 64, 0, stream>>>(acc);
    fps_kernel<<<NB, 256, 0, stream>>>(pcd, newxyz);
    repulsion_kernel<<<NB * (NPTS / 16), 32, 0, stream>>>(pcd, acc + 5);

    const dim3 ggrid(NB * NPOINT);
    group_kernel<16, 1><<<ggrid, 32, 0, stream>>>(pcd, newxyz, acc + 0, 0.004f);
    group_kernel<24, 2><<<ggrid, 32, 0, stream>>>(pcd, newxyz, acc + 1, 0.006f);
    group_kernel<32, 2><<<ggrid, 32, 0, stream>>>(pcd, newxyz, acc + 2, 0.008f);
    group_kernel<40, 3><<<ggrid, 32, 0, stream>>>(pcd, newxyz, acc + 3, 0.010f);
    group_kernel<49, 4><<<ggrid, 32, 0, stream>>>(pcd, newxyz, acc + 4, 0.012f);

    final_kernel<<<1, 32, 0, stream>>>(acc, out);
}